// GCLModel_Morph_27479200759915
// MI455X (gfx1250) — compile-verified
//
#include <hip/hip_runtime.h>
#include <math.h>

typedef __attribute__((ext_vector_type(2))) float v2f;
typedef __attribute__((ext_vector_type(8))) float v8f;

#define NNODES 50000
#define NEDGES 800000

// ---------------------------------------------------------------------------
// Fused feature projection: Linear(15,64)+LN+GELU -> Linear(64,128)+LN+GELU
// One wave (32 lanes) per node; weights staged in LDS.
// ---------------------------------------------------------------------------
__global__ __launch_bounds__(256) void feature_proj_kernel(
    const float* __restrict__ x,
    const float* __restrict__ w1, const float* __restrict__ b1,
    const float* __restrict__ g1, const float* __restrict__ be1,
    const float* __restrict__ w2, const float* __restrict__ b2,
    const float* __restrict__ g2, const float* __restrict__ be2,
    float* __restrict__ h, int N)
{
    __shared__ float sW1[15 * 64];
    __shared__ float sW2[64 * 128];
    __shared__ float sB1[64], sG1[64], sBe1[64];
    __shared__ float sB2[128], sG2[128], sBe2[128];
    int t = threadIdx.x;
    for (int i = t; i < 15 * 64; i += 256) sW1[i] = w1[i];
    for (int i = t; i < 64 * 128; i += 256) sW2[i] = w2[i];
    if (t < 64) { sB1[t] = b1[t]; sG1[t] = g1[t]; sBe1[t] = be1[t]; }
    if (t < 128) { sB2[t] = b2[t]; sG2[t] = g2[t]; sBe2[t] = be2[t]; }
    __syncthreads();

    int wave = t >> 5, lane = t & 31;
    int node = blockIdx.x * 8 + wave;
    if (node >= N) return;

    float xv = (lane < 15) ? x[node * 15 + lane] : 0.f;
    float h0 = sB1[lane], h1 = sB1[lane + 32];
    #pragma unroll
    for (int k = 0; k < 15; ++k) {
        float xk = __shfl(xv, k);
        h0 += xk * sW1[k * 64 + lane];
        h1 += xk * sW1[k * 64 + lane + 32];
    }
    // LayerNorm over 64
    float s = h0 + h1;
    for (int o = 16; o; o >>= 1) s += __shfl_xor(s, o);
    float mu = s * (1.f / 64.f);
    float d0 = h0 - mu, d1 = h1 - mu;
    float vs = d0 * d0 + d1 * d1;
    for (int o = 16; o; o >>= 1) vs += __shfl_xor(vs, o);
    float inv = rsqrtf(vs * (1.f / 64.f) + 1e-5f);
    float n0 = d0 * inv * sG1[lane] + sBe1[lane];
    float n1 = d1 * inv * sG1[lane + 32] + sBe1[lane + 32];
    // exact GELU
    n0 = 0.5f * n0 * (1.f + erff(n0 * 0.70710678118f));
    n1 = 0.5f * n1 * (1.f + erff(n1 * 0.70710678118f));

    float a0 = sB2[lane], a1 = sB2[lane + 32], a2 = sB2[lane + 64], a3 = sB2[lane + 96];
    for (int k = 0; k < 64; ++k) {
        float src = (k < 32) ? n0 : n1;      // uniform condition
        float hk = __shfl(src, k & 31);
        const float* wr = &sW2[k * 128];
        a0 += hk * wr[lane];
        a1 += hk * wr[lane + 32];
        a2 += hk * wr[lane + 64];
        a3 += hk * wr[lane + 96];
    }
    // LayerNorm over 128
    float s2 = a0 + a1 + a2 + a3;
    for (int o = 16; o; o >>= 1) s2 += __shfl_xor(s2, o);
    float mu2 = s2 * (1.f / 128.f);
    float e0 = a0 - mu2, e1 = a1 - mu2, e2 = a2 - mu2, e3 = a3 - mu2;
    float v2 = e0 * e0 + e1 * e1 + e2 * e2 + e3 * e3;
    for (int o = 16; o; o >>= 1) v2 += __shfl_xor(v2, o);
    float inv2 = rsqrtf(v2 * (1.f / 128.f) + 1e-5f);
    float r0 = e0 * inv2 * sG2[lane] + sBe2[lane];
    float r1 = e1 * inv2 * sG2[lane + 32] + sBe2[lane + 32];
    float r2 = e2 * inv2 * sG2[lane + 64] + sBe2[lane + 64];
    float r3 = e3 * inv2 * sG2[lane + 96] + sBe2[lane + 96];
    r0 = 0.5f * r0 * (1.f + erff(r0 * 0.70710678118f));
    r1 = 0.5f * r1 * (1.f + erff(r1 * 0.70710678118f));
    r2 = 0.5f * r2 * (1.f + erff(r2 * 0.70710678118f));
    r3 = 0.5f * r3 * (1.f + erff(r3 * 0.70710678118f));
    float* out = h + (long)node * 128;
    out[lane] = r0; out[lane + 32] = r1; out[lane + 64] = r2; out[lane + 96] = r3;
}

// ---------------------------------------------------------------------------
// f32 WMMA GEMM: C[M,N] = A[M,K] * B[K,N] (+bias) (+accumulate into C) (ReLU)
// One wave per 16x16 tile. M%16==0, K%4==0, N%16==0. EXEC all-ones per wave.
// ---------------------------------------------------------------------------
template <bool BIAS, bool RELU, bool ACCUM>
__global__ __launch_bounds__(256) void gemm_wmma_f32(
    const float* __restrict__ A, const float* __restrict__ B,
    const float* __restrict__ bias, float* __restrict__ C,
    int M, int K, int Ncols, int tilesN)
{
    int wid = (int)((blockIdx.x * blockDim.x + threadIdx.x) >> 5);
    int lane = threadIdx.x & 31;
    int tilesM = M >> 4;
    if (wid >= tilesM * tilesN) return;           // wave-uniform exit
    int mt = wid / tilesN, nt = wid % tilesN;
    int m0 = mt << 4, n0 = nt << 4;
    int hf = lane >> 4;                           // half-wave select
    int l16 = lane & 15;

    v8f acc = {};
    if (ACCUM) {
        #pragma unroll
        for (int r = 0; r < 8; ++r)
            acc[r] = C[(long)(m0 + r + 8 * hf) * Ncols + n0 + l16];
    }

    const float* Arow = A + (long)(m0 + l16) * K;
    for (int k0 = 0; k0 < K; k0 += 4) {
        // A 16x4 f32 fragment: lanes<16 hold K0/K1, lanes>=16 hold K2/K3
        v2f a = *(const v2f*)(Arow + k0 + 2 * hf);
        // B 4x16 f32 fragment: row = vgpr + 2*half, col = lane%16
        v2f b;
        b.x = B[(long)(k0 + 2 * hf) * Ncols + n0 + l16];
        b.y = B[(long)(k0 + 2 * hf + 1) * Ncols + n0 + l16];
        acc = __builtin_amdgcn_wmma_f32_16x16x4_f32(
            false, a, false, b, (short)0, acc, false, false);
    }

    float bv = BIAS ? bias[n0 + l16] : 0.f;
    #pragma unroll
    for (int r = 0; r < 8; ++r) {
        float v = acc[r] + bv;
        if (RELU) v = v > 0.f ? v : 0.f;
        C[(long)(m0 + r + 8 * hf) * Ncols + n0 + l16] = v;
    }
}

// ---------------------------------------------------------------------------
// Attention-score prep: a_src[n,h] = sum_c hh[n,h,c]*att_src[h,c] (C=128)
// One wave per node.
// ---------------------------------------------------------------------------
__global__ __launch_bounds__(256) void attn_prep_kernel(
    const float* __restrict__ hh, const float* __restrict__ att_src,
    const float* __restrict__ att_dst, float* __restrict__ asrc,
    float* __restrict__ adst, int N, int H)
{
    int wid = (int)((blockIdx.x * blockDim.x + threadIdx.x) >> 5);
    int lane = threadIdx.x & 31;
    if (wid >= N) return;
    const float* row = hh + (long)wid * H * 128;
    for (int hh_i = 0; hh_i < H; ++hh_i) {
        float s1 = 0.f, s2 = 0.f;
        #pragma unroll
        for (int c = lane; c < 128; c += 32) {
            float v = row[hh_i * 128 + c];
            s1 += v * att_src[hh_i * 128 + c];
            s2 += v * att_dst[hh_i * 128 + c];
        }
        for (int o = 16; o; o >>= 1) { s1 += __shfl_xor(s1, o); s2 += __shfl_xor(s2, o); }
        if (lane == 0) { asrc[wid * H + hh_i] = s1; adst[wid * H + hh_i] = s2; }
    }
}

__global__ void fill_f32_kernel(float* p, float v, int n) {
    int i = blockIdx.x * blockDim.x + threadIdx.x;
    if (i < n) p[i] = v;
}

__device__ inline void atomicMaxF(float* addr, float val) {
    unsigned int* ua = (unsigned int*)addr;
    unsigned int old = __float_as_uint(*addr);
    while (__uint_as_float(old) < val) {
        unsigned int assumed = old;
        old = atomicCAS(ua, assumed, __float_as_uint(val));
        if (old == assumed) break;
    }
}

// pass A: per-edge logits + segment max over dst
__global__ void edge_logits_max_kernel(
    const long long* __restrict__ src, const long long* __restrict__ dst,
    const float* __restrict__ ea, const float* __restrict__ asrc,
    const float* __restrict__ adst, const float* __restrict__ lam,
    float* __restrict__ alpha, float* __restrict__ amax, int E, int H)
{
    int e = blockIdx.x * blockDim.x + threadIdx.x;
    if (e >= E) return;
    int s = (int)src[e], d = (int)dst[e];
    float lg = lam[0] * logf(ea[e] + 1e-6f);
    for (int h = 0; h < H; ++h) {
        float a = asrc[d * H + h] + adst[s * H + h];
        a = a > 0.f ? a : 0.2f * a;           // leaky_relu(0.2)
        a += lg;
        alpha[(long)e * H + h] = a;
        atomicMaxF(&amax[d * H + h], a);
    }
}

// pass B: exponentiate and accumulate denominator
__global__ void edge_exp_sum_kernel(
    const long long* __restrict__ dst, float* __restrict__ alpha,
    const float* __restrict__ amax, float* __restrict__ denom, int E, int H)
{
    int e = blockIdx.x * blockDim.x + threadIdx.x;
    if (e >= E) return;
    int d = (int)dst[e];
    for (int h = 0; h < H; ++h) {
        float w = __expf(alpha[(long)e * H + h] - amax[d * H + h]);
        alpha[(long)e * H + h] = w;
        atomicAdd(&denom[d * H + h], w);
    }
}

// pass C: weighted scatter of hh[src] into agg[dst].  grid = E blocks,
// block = H*128/4 threads, float4 per thread.  C per head is 128.
__global__ void edge_scatter_kernel(
    const long long* __restrict__ src, const long long* __restrict__ dst,
    const float* __restrict__ alpha, const float* __restrict__ denom,
    const float* __restrict__ hh, float* __restrict__ agg, int H)
{
    int e = blockIdx.x;
    int c4 = threadIdx.x * 4;
    int h = c4 >> 7;                          // C == 128
    int s = (int)src[e], d = (int)dst[e];
    float w = alpha[(long)e * H + h] / (denom[d * H + h] + 1e-16f);
    float4 v = *(const float4*)(hh + (long)s * H * 128 + c4);
    float* o = agg + (long)d * H * 128 + c4;
    atomicAdd(o + 0, v.x * w);
    atomicAdd(o + 1, v.y * w);
    atomicAdd(o + 2, v.z * w);
    atomicAdd(o + 3, v.w * w);
}

// ELU(x + bias) in place (F is a power of two)
__global__ void elu_bias_kernel(float* agg, const float* __restrict__ bias, int total, int Fmask) {
    int i = blockIdx.x * blockDim.x + threadIdx.x;
    if (i >= total) return;
    float v = agg[i] + bias[i & Fmask];
    agg[i] = v > 0.f ? v : expm1f(v);
}

// out = agg + bias
__global__ void add_bias_out_kernel(const float* __restrict__ agg,
                                    const float* __restrict__ bias,
                                    float* __restrict__ out, int total, int Fmask) {
    int i = blockIdx.x * blockDim.x + threadIdx.x;
    if (i >= total) return;
    out[i] = agg[i] + bias[i & Fmask];
}

// ---------------------------------------------------------------------------
extern "C" void kernel_launch(void* const* d_in, const int* in_sizes, int n_in,
                              void* d_out, int out_size, void* d_ws, size_t ws_size,
                              hipStream_t stream)
{
    const int N = NNODES, E = NEDGES;
    const float*     x      = (const float*)d_in[0];
    const long long* ei     = (const long long*)d_in[1];
    const long long* src    = ei;
    const long long* dst    = ei + E;
    const float*     ea     = (const float*)d_in[2];
    const float* fp_w1 = (const float*)d_in[3];
    const float* fp_b1 = (const float*)d_in[4];
    const float* fp_g1 = (const float*)d_in[5];
    const float* fp_be1= (const float*)d_in[6];
    const float* fp_w2 = (const float*)d_in[7];
    const float* fp_b2 = (const float*)d_in[8];
    const float* fp_g2 = (const float*)d_in[9];
    const float* fp_be2= (const float*)d_in[10];
    const float* c1_lin     = (const float*)d_in[11];
    const float* c1_att_src = (const float*)d_in[12];
    const float* c1_att_dst = (const float*)d_in[13];
    const float* c1_bias    = (const float*)d_in[14];
    const float* c1_lambda  = (const float*)d_in[15];
    const float* skip_w     = (const float*)d_in[16];
    const float* skip_b     = (const float*)d_in[17];
    const float* c2_lin     = (const float*)d_in[18];
    const float* c2_att_src = (const float*)d_in[19];
    const float* c2_att_dst = (const float*)d_in[20];
    const float* c2_bias    = (const float*)d_in[21];
    const float* c2_lambda  = (const float*)d_in[22];
    const float* ph_w1      = (const float*)d_in[23];
    const float* ph_b1      = (const float*)d_in[24];
    const float* ph_w2      = (const float*)d_in[25];
    const float* ph_b2      = (const float*)d_in[26];

    float* ws = (float*)d_ws;
    size_t o = 0;
    float* h     = ws + o; o += (size_t)N * 128;   // x_in
    float* hh1   = ws + o; o += (size_t)N * 512;   // conv1 transformed feats
    float* agg1  = ws + o; o += (size_t)N * 512;   // conv1 aggregate -> h2
    float* asrc1 = ws + o; o += (size_t)N * 4;
    float* adst1 = ws + o; o += (size_t)N * 4;
    float* amax1 = ws + o; o += (size_t)N * 4;
    float* den1  = ws + o; o += (size_t)N * 4;
    float* alpha1= ws + o; o += (size_t)E * 4;
    float* asrc2 = ws + o; o += (size_t)N;
    float* adst2 = ws + o; o += (size_t)N;
    float* amax2 = ws + o; o += (size_t)N;
    float* den2  = ws + o; o += (size_t)N;
    float* alpha2= ws + o; o += (size_t)E;
    float* hh2  = hh1;   // reuse: hh1 dead after conv1 scatter
    float* agg2 = agg1;  // reuse: h2 dead after conv2 GEMM
    float* tbuf = h;     // reuse: h dead after skip GEMM

    float* zout     = (float*)d_out;                 // [N,32]
    float* node_emb = (float*)d_out + (size_t)N * 32; // [N,128]

    const int TPB = 256;
    auto gemm_blocks = [](int M, int Ncols) {
        int waves = (M >> 4) * (Ncols >> 4);
        return (waves + 7) / 8;
    };

    // 1. feature projection -> h [N,128]
    feature_proj_kernel<<<(N + 7) / 8, TPB, 0, stream>>>(
        x, fp_w1, fp_b1, fp_g1, fp_be1, fp_w2, fp_b2, fp_g2, fp_be2, h, N);

    // 2. hh1 = h @ c1_lin  [N,512]
    gemm_wmma_f32<false, false, false><<<gemm_blocks(N, 512), TPB, 0, stream>>>(
        h, c1_lin, nullptr, hh1, N, 128, 512, 512 / 16);

    // 3. attention scores (4 heads)
    attn_prep_kernel<<<(N * 32 + TPB - 1) / TPB, TPB, 0, stream>>>(
        hh1, c1_att_src, c1_att_dst, asrc1, adst1, N, 4);

    // 4. init segment-softmax state + aggregate buffer
    fill_f32_kernel<<<(N * 4 + TPB - 1) / TPB, TPB, 0, stream>>>(amax1, -3.0e38f, N * 4);
    fill_f32_kernel<<<(N * 4 + TPB - 1) / TPB, TPB, 0, stream>>>(den1, 0.f, N * 4);
    fill_f32_kernel<<<(N * 512 + TPB - 1) / TPB, TPB, 0, stream>>>(agg1, 0.f, N * 512);

    // 5-7. conv1 edge passes
    edge_logits_max_kernel<<<(E + TPB - 1) / TPB, TPB, 0, stream>>>(
        src, dst, ea, asrc1, adst1, c1_lambda, alpha1, amax1, E, 4);
    edge_exp_sum_kernel<<<(E + TPB - 1) / TPB, TPB, 0, stream>>>(
        dst, alpha1, amax1, den1, E, 4);
    edge_scatter_kernel<<<E, 128, 0, stream>>>(
        src, dst, alpha1, den1, hh1, agg1, 4);

    // 8. h2 = elu(agg1 + c1_bias) ... (in place)
    elu_bias_kernel<<<(N * 512 + TPB - 1) / TPB, TPB, 0, stream>>>(
        agg1, c1_bias, N * 512, 511);
    // 9. ... + h @ skip_w + skip_b   (accumulate GEMM into agg1 -> h2)
    gemm_wmma_f32<true, false, true><<<gemm_blocks(N, 512), TPB, 0, stream>>>(
        h, skip_w, skip_b, agg1, N, 128, 512, 512 / 16);

    // 10. hh2 = h2 @ c2_lin  [N,128]
    gemm_wmma_f32<false, false, false><<<gemm_blocks(N, 128), TPB, 0, stream>>>(
        agg1, c2_lin, nullptr, hh2, N, 512, 128, 128 / 16);

    // 11. attention scores (1 head)
    attn_prep_kernel<<<(N * 32 + TPB - 1) / TPB, TPB, 0, stream>>>(
        hh2, c2_att_src, c2_att_dst, asrc2, adst2, N, 1);

    // 12. init layer-2 softmax state + aggregate
    fill_f32_kernel<<<(N + TPB - 1) / TPB, TPB, 0, stream>>>(amax2, -3.0e38f, N);
    fill_f32_kernel<<<(N + TPB - 1) / TPB, TPB, 0, stream>>>(den2, 0.f, N);
    fill_f32_kernel<<<(N * 128 + TPB - 1) / TPB, TPB, 0, stream>>>(agg2, 0.f, N * 128);

    // 13-15. conv2 edge passes
    edge_logits_max_kernel<<<(E + TPB - 1) / TPB, TPB, 0, stream>>>(
        src, dst, ea, asrc2, adst2, c2_lambda, alpha2, amax2, E, 1);
    edge_exp_sum_kernel<<<(E + TPB - 1) / TPB, TPB, 0, stream>>>(
        dst, alpha2, amax2, den2, E, 1);
    edge_scatter_kernel<<<E, 32, 0, stream>>>(
        src, dst, alpha2, den2, hh2, agg2, 1);

    // 16. node_emb = agg2 + c2_bias   (second output)
    add_bias_out_kernel<<<(N * 128 + TPB - 1) / TPB, TPB, 0, stream>>>(
        agg2, c2_bias, node_emb, N * 128, 127);

    // 17. t = relu(node_emb @ ph_w1 + ph_b1)
    gemm_wmma_f32<true, true, false><<<gemm_blocks(N, 128), TPB, 0, stream>>>(
        node_emb, ph_w1, ph_b1, tbuf, N, 128, 128, 128 / 16);
    // 18. z = t @ ph_w2 + ph_b2       (first output)
    gemm_wmma_f32<true, false, false><<<gemm_blocks(N, 32), TPB, 0, stream>>>(
        tbuf, ph_w2, ph_b2, zout, N, 128, 32, 32 / 16);
}